// PyramidGrid_61065845014889
// MI455X (gfx1250) — compile-verified
//
#include <hip/hip_runtime.h>
#include <hip/hip_bf16.h>

// Multiresolution simplex hash-grid encoding forward pass for MI455X (gfx1250).
//
// Thread mapping: tid = b*16 + l  (b = point, l = level).
//  - 16 consecutive lanes = 16 levels of one point -> one wave32 stores 256
//    contiguous bytes of output (coalesced global_store_b64, non-temporal).
//  - Inputs are staged into LDS with the CDNA5 async global->LDS path.
//
// Hash sizes per level are powers of two (4096, 32768, 262144, 524288...),
// so modulo is a mask. Ranks reproduce jnp stable argsort tie-breaking.

#define NLV   16
#define BLOCK 256
#define PPB   (BLOCK / NLV)   // 16 points per block

typedef __attribute__((address_space(3))) float* lds_f32p;
typedef float v2f __attribute__((ext_vector_type(2)));

__global__ __launch_bounds__(BLOCK) void pyramid_hashgrid_fwd(
    const float* __restrict__ inputs,      // [B, 3]
    const float* __restrict__ embeddings,  // [7114752, 2]
    float* __restrict__ out,               // [B, 32]
    int batch)
{
    __shared__ float s_pts[PPB * 3];  // 48 floats = 192 B

    const int tid     = threadIdx.x;
    const int base_pt = blockIdx.x * PPB;

    // ---- Stage this block's 16 input points into LDS via async DMA ----
    if (tid < PPB * 3) {
        const int gidx = base_pt * 3 + tid;
        if (gidx < batch * 3) {
            unsigned goff    = (unsigned)gidx * 4u;                       // byte offset from base
            unsigned lds_off = (unsigned)(size_t)(lds_f32p)(&s_pts[tid]); // LDS byte address
            asm volatile("global_load_async_to_lds_b32 %0, %1, %2"
                         :: "v"(lds_off), "v"(goff), "s"(inputs)
                         : "memory");
        }
    }
    asm volatile("s_wait_asynccnt 0x0" ::: "memory");
    __syncthreads();

    const int p = tid >> 4;          // point within block
    const int l = tid & (NLV - 1);   // level
    const int b = base_pt + p;
    if (b >= batch) return;

    const float in0 = s_pts[p * 3 + 0];
    const float in1 = s_pts[p * 3 + 1];
    const float in2 = s_pts[p * 3 + 2];

    // ---- Per-level constants (all hash sizes are powers of two) ----
    unsigned mask, prev;
    if (l >= 3)      { mask = 524287u; prev = 299008u + (unsigned)(l - 3) * 524288u; }
    else if (l == 2) { mask = 262143u; prev = 36864u; }
    else if (l == 1) { mask = 32767u;  prev = 4096u;  }
    else             { mask = 4095u;   prev = 0u;     }

    const float fres = (float)(16 << l);   // exact: 16..524288

    const float x0 = in0 * fres;
    const float x1 = in1 * fres;
    const float x2 = in2 * fres;
    const int  xi0 = (int)x0;              // trunc == floor (x >= 0)
    const int  xi1 = (int)x1;
    const int  xi2 = (int)x2;
    const float f0 = x0 - (float)xi0;
    const float f1 = x1 - (float)xi1;
    const float f2 = x2 - (float)xi2;

    // Stable-argsort ranks: rank_j = #{m: f_m < f_j} + #{m<j: f_m == f_j}
    const int r0 = (int)(f1 < f0)  + (int)(f2 < f0);
    const int r1 = (int)(f0 <= f1) + (int)(f2 < f1);
    const int r2 = (int)(f0 <= f2) + (int)(f1 <= f2);

    // Sorted fractional values via exact min/median/max network
    const float s0 = fminf(fminf(f0, f1), f2);
    const float s2 = fmaxf(fmaxf(f0, f1), f2);
    const float s1 = fmaxf(fminf(f0, f1), fminf(fmaxf(f0, f1), f2));

    // ---- Hash the 4 simplex corners: corner k adds 1 where rank_j >= k ----
    const unsigned P1 = 2654435761u, P2 = 805459861u;
    unsigned idx[4];
#pragma unroll
    for (int k = 0; k < 4; ++k) {
        const unsigned c0 = (unsigned)(xi0 + (int)(r0 >= k));
        const unsigned c1 = (unsigned)(xi1 + (int)(r1 >= k));
        const unsigned c2 = (unsigned)(xi2 + (int)(r2 >= k));
        const unsigned h  = c0 ^ (c1 * P1) ^ (c2 * P2);
        idx[k] = (h & mask) + prev;
    }

    // ---- Gather 4 embedding rows (8 B each; table is L2-resident) ----
    const float2* __restrict__ emb2 = (const float2*)embeddings;
    const float2 e0 = emb2[idx[0]];
    const float2 e1 = emb2[idx[1]];
    const float2 e2 = emb2[idx[2]];
    const float2 e3 = emb2[idx[3]];

    // ---- Barycentric blend: w = [s0, s1-s0, s2-s1, 1-s2] ----
    const float w0 = s0;
    const float w1 = s1 - s0;
    const float w2 = s2 - s1;
    const float w3 = 1.0f - s2;

    v2f o;
    o.x = w0 * e0.x + w1 * e1.x + w2 * e2.x + w3 * e3.x;
    o.y = w0 * e0.y + w1 * e1.y + w2 * e2.y + w3 * e3.y;

    // Streaming output (256 MB written once): non-temporal so the 57 MB
    // embedding table stays resident in the 192 MB L2.
    v2f* __restrict__ out2 = (v2f*)out;
    __builtin_nontemporal_store(o, &out2[(size_t)b * NLV + l]);
}

extern "C" void kernel_launch(void* const* d_in, const int* in_sizes, int n_in,
                              void* d_out, int out_size, void* d_ws, size_t ws_size,
                              hipStream_t stream) {
    const float* inputs     = (const float*)d_in[0];  // [B,3] float32
    const float* embeddings = (const float*)d_in[1];  // [7114752,2] float32
    float* out              = (float*)d_out;          // [B,32] float32

    const int batch  = in_sizes[0] / 3;
    const int blocks = (batch + PPB - 1) / PPB;       // 16 points per 256-thread block

    pyramid_hashgrid_fwd<<<blocks, BLOCK, 0, stream>>>(inputs, embeddings, out, batch);
}